// Clause_Hypergraph_25254407701308
// MI455X (gfx1250) — compile-verified
//
#include <hip/hip_runtime.h>
#include <hip/hip_bf16.h>
#include <math.h>

// ---------------- problem constants (match reference) ----------------
#define BS    8
#define N_PER 8192
#define NTOT  (BS * N_PER)      // 65536 nodes
#define EMB   128
#define ENCD  512
#define LSEQ  512
#define ROWS_X (BS * LSEQ)      // 4096
#define E_TOT 1048576           // BS * 131072 edges
#define P_TOT 1048576           // incidence pairs
#define N_HE  4096
#define CATD  (ENCD + EMB)      // 640
#define OUTD  256

#define KC    16                // GEMM k-chunk
#define APAD  20                // A-tile row stride (16B aligned, bank-conflict free)

typedef float v2f __attribute__((ext_vector_type(2)));
typedef float v8f __attribute__((ext_vector_type(8)));

static __device__ __forceinline__ v8f wmma4(v2f a, v2f b, v8f c) {
  // D = A(16x4,f32) * B(4x16,f32) + C(16x16,f32)
  return __builtin_amdgcn_wmma_f32_16x16x4_f32(false, a, false, b, (short)0, c,
                                               false, false);
}

// --------- CDNA5 async global->LDS copy (ASYNCcnt-tracked DMA path) ---------
typedef __attribute__((address_space(3))) char lds_char_t;

static __device__ __forceinline__ void async_copy_b128(void* lds_ptr,
                                                       const void* gptr) {
  uint32_t loff = (uint32_t)(uint64_t)(lds_char_t*)lds_ptr;  // raw LDS byte offset
  uint64_t g = (uint64_t)gptr;
  asm volatile("global_load_async_to_lds_b128 %0, %1, off"
               :
               : "v"(loff), "v"(g)
               : "memory");
}

static __device__ __forceinline__ void wait_async_le4() {
  asm volatile("s_wait_asynccnt 0x4" ::: "memory");
}
static __device__ __forceinline__ void wait_async_zero() {
  asm volatile("s_wait_asynccnt 0x0" ::: "memory");
}

// ---------------- utility kernels ----------------
__global__ void zero_f32(float* __restrict__ p, long n) {
  long i = (long)blockIdx.x * blockDim.x + threadIdx.x;
  if (i < n) p[i] = 0.0f;
}

__global__ void deg_edges(const int* __restrict__ src, const int* __restrict__ dst,
                          float* __restrict__ dout, float* __restrict__ din, int nE) {
  int e = blockIdx.x * blockDim.x + threadIdx.x;
  if (e >= nE) return;
  atomicAdd(&dout[src[e]], 1.0f);
  atomicAdd(&din[dst[e]], 1.0f);
}

__global__ void deg_pairs(const int* __restrict__ pairs, float* __restrict__ D,
                          float* __restrict__ B, int nP) {
  int p = blockIdx.x * blockDim.x + threadIdx.x;
  if (p >= nP) return;
  atomicAdd(&D[pairs[2 * p + 0]], 1.0f);
  atomicAdd(&B[pairs[2 * p + 1]], 1.0f);
}

__global__ void rsqrt_clamp(float* __restrict__ a, int n) {
  int i = blockIdx.x * blockDim.x + threadIdx.x;
  if (i < n) a[i] = rsqrtf(fmaxf(a[i], 1.0f));
}

__global__ void invert_pos(float* __restrict__ a, int n) {
  int i = blockIdx.x * blockDim.x + threadIdx.x;
  if (i < n) { float v = a[i]; a[i] = (v > 0.0f) ? (1.0f / v) : 0.0f; }
}

// cw[c] = sum_k cb[k] * W[k][c]   (folds input column-bias through the GEMM)
__global__ void colbias_proj(const float* __restrict__ cb, const float* __restrict__ W,
                             int K, float* __restrict__ cw) {
  int c = threadIdx.x;
  float s = 0.0f;
  for (int k = 0; k < K; ++k) s += cb[k] * W[(size_t)k * 128 + c];
  cw[c] = s;
}

// agg[dst] += t[src]   (one wave per edge, float4 per lane)
__global__ void scatter_edges(const float* __restrict__ t, float* __restrict__ agg,
                              const int* __restrict__ src, const int* __restrict__ dst,
                              int nE) {
  int idx = blockIdx.x * blockDim.x + threadIdx.x;
  int e = idx >> 5;
  if (e >= nE) return;
  int c = (idx & 31) << 2;
  int s = src[e], d = dst[e];
  const float4 v = *(const float4*)&t[(size_t)s * EMB + c];
  float* o = &agg[(size_t)d * EMB + c];
  atomicAdd(o + 0, v.x); atomicAdd(o + 1, v.y);
  atomicAdd(o + 2, v.z); atomicAdd(o + 3, v.w);
}

// ef[pe] += t[pn]
__global__ void scatter_pairs_fwd(const float* __restrict__ t, float* __restrict__ ef,
                                  const int* __restrict__ pairs, int nP) {
  int idx = blockIdx.x * blockDim.x + threadIdx.x;
  int p = idx >> 5;
  if (p >= nP) return;
  int c = (idx & 31) << 2;
  int pn = pairs[2 * p + 0], pe = pairs[2 * p + 1];
  const float4 v = *(const float4*)&t[(size_t)pn * EMB + c];
  float* o = &ef[(size_t)pe * EMB + c];
  atomicAdd(o + 0, v.x); atomicAdd(o + 1, v.y);
  atomicAdd(o + 2, v.z); atomicAdd(o + 3, v.w);
}

// nagg[pn] += Binv[pe] * ef[pe]
__global__ void scatter_pairs_bwd(const float* __restrict__ ef, const float* __restrict__ Binv,
                                  float* __restrict__ nagg, const int* __restrict__ pairs,
                                  int nP) {
  int idx = blockIdx.x * blockDim.x + threadIdx.x;
  int p = idx >> 5;
  if (p >= nP) return;
  int c = (idx & 31) << 2;
  int pn = pairs[2 * p + 0], pe = pairs[2 * p + 1];
  float s = Binv[pe];
  const float4 v = *(const float4*)&ef[(size_t)pe * EMB + c];
  float* o = &nagg[(size_t)pn * EMB + c];
  atomicAdd(o + 0, s * v.x); atomicAdd(o + 1, s * v.y);
  atomicAdd(o + 2, s * v.z); atomicAdd(o + 3, s * v.w);
}

// h = relu(hagg * Dinv[row] + bh[col]) + nfr
__global__ void finalize_h(const float* __restrict__ hagg, const float* __restrict__ Dinv,
                           const float* __restrict__ bh, const float* __restrict__ nfr,
                           float* __restrict__ hout, long n) {
  long i = (long)blockIdx.x * blockDim.x + threadIdx.x;
  if (i >= n) return;
  int row = (int)(i >> 7);
  int col = (int)(i & 127);
  float v = hagg[i] * Dinv[row] + bh[col];
  hout[i] = fmaxf(v, 0.0f) + nfr[i];
}

// cat = [x | Hatt]
__global__ void concat_xh(const float* __restrict__ x, const float* __restrict__ H,
                          float* __restrict__ cat) {
  long i = (long)blockIdx.x * blockDim.x + threadIdx.x;
  long n = (long)ROWS_X * CATD;
  if (i >= n) return;
  int r = (int)(i / CATD);
  int c = (int)(i % CATD);
  cat[i] = (c < ENCD) ? x[(size_t)r * ENCD + c] : H[(size_t)r * EMB + (c - ENCD)];
}

// ---------------- WMMA f32 GEMM with async-DMA double-buffered tiles --------
// Y = act( srow[r] * (X @ W) + cw[c] + bias[c] )
// X: [rows,ldx] uses cols [0,K) (rows % 128 == 0, K % 16 == 0); W: [K,128].
// mode: 0=none 1=relu 2=sigmoid 3=tanh
static __device__ __forceinline__ void issue_tiles(
    const float* __restrict__ X, int ldx, const float* __restrict__ W,
    int rowbase, int kc, int tid, float (*As)[APAD], float (*Bs)[128]) {
#pragma unroll
  for (int i = 0; i < 2; ++i) {           // A tile: 128 x 16 floats
    int chunk = tid + i * 256;            // 512 x B128 chunks
    int r = chunk >> 2, c4 = (chunk & 3) << 2;
    async_copy_b128(&As[r][c4], &X[(size_t)(rowbase + r) * ldx + kc + c4]);
  }
#pragma unroll
  for (int i = 0; i < 2; ++i) {           // B tile: 16 x 128 floats
    int chunk = tid + i * 256;
    int r = chunk >> 5, c4 = (chunk & 31) << 2;
    async_copy_b128(&Bs[r][c4], &W[(size_t)(kc + r) * 128 + c4]);
  }
}

__global__ __launch_bounds__(256) void gemm128(
    const float* __restrict__ X, int ldx, int rows, int K,
    const float* __restrict__ W,
    const float* __restrict__ srow, const float* __restrict__ cw,
    const float* __restrict__ bias,
    float* __restrict__ Y, int ldy, int ycol, int mode) {
  __shared__ float As[2][128][APAD];   // 20.0 KB
  __shared__ float Bs[2][KC][128];     // 16.0 KB

  const int tid = threadIdx.x;
  const int wv = tid >> 5;
  const int lane = tid & 31;
  const int lm = lane & 15;
  const int hi = (lane >= 16) ? 1 : 0;
  const int rowbase = blockIdx.x * 128;
  const int nchunks = K / KC;

  v8f acc[8];
  const v8f vz = {0.f, 0.f, 0.f, 0.f, 0.f, 0.f, 0.f, 0.f};
#pragma unroll
  for (int i = 0; i < 8; ++i) acc[i] = vz;

  // prologue: start DMA of chunk 0
  issue_tiles(X, ldx, W, rowbase, 0, tid, As[0], Bs[0]);

  for (int ch = 0; ch < nchunks; ++ch) {
    const int buf = ch & 1;
    if (ch + 1 < nchunks) {
      issue_tiles(X, ldx, W, rowbase, (ch + 1) * KC, tid, As[buf ^ 1], Bs[buf ^ 1]);
      wait_async_le4();   // retire chunk ch's group (4 async ops/wave in flight max)
    } else {
      wait_async_zero();
    }
    __syncthreads();

#pragma unroll
    for (int ks = 0; ks < KC / 4; ++ks) {
      int koff = ks * 4 + (hi ? 2 : 0);
      v2f a;
      a.x = As[buf][wv * 16 + lm][koff];
      a.y = As[buf][wv * 16 + lm][koff + 1];
#pragma unroll
      for (int ct = 0; ct < 8; ++ct) {
        int nn = ct * 16 + lm;
        v2f b;
        b.x = Bs[buf][koff][nn];
        b.y = Bs[buf][koff + 1][nn];
        acc[ct] = wmma4(a, b, acc[ct]);
      }
    }
    __syncthreads();
  }

  // epilogue: row-scale + column adds + activation
#pragma unroll
  for (int ct = 0; ct < 8; ++ct) {
    int col = ct * 16 + lm;
    float add = (cw ? cw[col] : 0.0f) + (bias ? bias[col] : 0.0f);
#pragma unroll
    for (int j = 0; j < 8; ++j) {
      int grow = rowbase + wv * 16 + j + (hi ? 8 : 0);
      if (grow < rows) {
        float v = acc[ct][j];
        if (srow) v *= srow[grow];
        v += add;
        if (mode == 1)      v = fmaxf(v, 0.0f);
        else if (mode == 2) v = 1.0f / (1.0f + __expf(-v));
        else if (mode == 3) v = tanhf(v);
        Y[(size_t)grow * ldy + ycol + col] = v;
      }
    }
  }
}

// ---------------- flash-attention readout ----------------
// per wave: 16 q-rows; stream all 8192 keys of the row's batch with online softmax.
__global__ __launch_bounds__(128) void attn_kernel(
    const float* __restrict__ q, const float* __restrict__ kbuf,
    const float* __restrict__ g, float* __restrict__ H) {
  __shared__ float qs[4][16][128];  // per-wave q tile
  __shared__ float ps[4][16][17];   // per-wave P tile (D-layout -> A-layout)

  const int tid = threadIdx.x;
  const int wv = tid >> 5;
  const int lane = tid & 31;
  const int lm = lane & 15;
  const int hi = (lane >= 16) ? 1 : 0;
  const int qr0 = blockIdx.x * 64 + wv * 16;     // global q row base
  const int b = qr0 >> 9;                        // / LSEQ
  const int nbase = b * N_PER;

  for (int i = lane; i < 16 * 128; i += 32) {
    int r = i >> 7, c = i & 127;
    qs[wv][r][c] = q[(size_t)(qr0 + r) * EMB + c];
  }
  __syncthreads();

  float m_[8], l_[8];
  v8f acc[8];
  const v8f vz = {0.f, 0.f, 0.f, 0.f, 0.f, 0.f, 0.f, 0.f};
#pragma unroll
  for (int j = 0; j < 8; ++j) { m_[j] = -1.0e30f; l_[j] = 0.0f; acc[j] = vz; }

  for (int nt = 0; nt < N_PER / 16; ++nt) {
    const int n0 = nbase + nt * 16;

    // S = q_tile @ k_tile^T   (16x16, K = 128 in 32 WMMA steps)
    v8f S = vz;
#pragma unroll
    for (int kf = 0; kf < 32; ++kf) {
      int koff = kf * 4 + (hi ? 2 : 0);
      v2f a;
      a.x = qs[wv][lm][koff];
      a.y = qs[wv][lm][koff + 1];
      const float* kp = &kbuf[(size_t)(n0 + lm) * EMB + koff];
      v2f bfr;
      bfr.x = kp[0];
      bfr.y = kp[1];
      S = wmma4(a, bfr, S);
    }

    // online softmax stats (rows live in VGPR index, cols across 16-lane halves)
    float mn[8], sc[8];
#pragma unroll
    for (int j = 0; j < 8; ++j) {
      float r = S[j];
      r = fmaxf(r, __shfl_xor(r, 1));
      r = fmaxf(r, __shfl_xor(r, 2));
      r = fmaxf(r, __shfl_xor(r, 4));
      r = fmaxf(r, __shfl_xor(r, 8));
      mn[j] = fmaxf(m_[j], r);
      sc[j] = __expf(m_[j] - mn[j]);
    }
#pragma unroll
    for (int j = 0; j < 8; ++j) {
      float p = __expf(S[j] - mn[j]);
      S[j] = p;
      float rs = p;
      rs += __shfl_xor(rs, 1);
      rs += __shfl_xor(rs, 2);
      rs += __shfl_xor(rs, 4);
      rs += __shfl_xor(rs, 8);
      l_[j] = l_[j] * sc[j] + rs;
      m_[j] = mn[j];
    }
#pragma unroll
    for (int dt = 0; dt < 8; ++dt)
#pragma unroll
      for (int j = 0; j < 8; ++j) acc[dt][j] *= sc[j];

    // P: D-layout -> LDS -> A-layout
#pragma unroll
    for (int j = 0; j < 8; ++j) ps[wv][j + (hi ? 8 : 0)][lm] = S[j];
    __syncthreads();

    // acc += P @ g_tile   (K = 16 in 4 WMMA steps, 8 d-tiles)
#pragma unroll
    for (int kf = 0; kf < 4; ++kf) {
      int koff = kf * 4 + (hi ? 2 : 0);
      v2f a;
      a.x = ps[wv][lm][koff];
      a.y = ps[wv][lm][koff + 1];
      const float* g0 = &g[(size_t)(n0 + koff) * EMB];
#pragma unroll
      for (int dt = 0; dt < 8; ++dt) {
        int dcol = dt * 16 + lm;
        v2f bfr;
        bfr.x = g0[dcol];
        bfr.y = g0[EMB + dcol];
        acc[dt] = wmma4(a, bfr, acc[dt]);
      }
    }
    __syncthreads();
  }

  // normalize and store
  float inv[8];
#pragma unroll
  for (int j = 0; j < 8; ++j) inv[j] = 1.0f / l_[j];
#pragma unroll
  for (int dt = 0; dt < 8; ++dt)
#pragma unroll
    for (int j = 0; j < 8; ++j) {
      int grow = qr0 + j + (hi ? 8 : 0);
      H[(size_t)grow * EMB + dt * 16 + lm] = acc[dt][j] * inv[j];
    }
}

// ---------------- host-side orchestration ----------------
static inline int cdiv(long a, int b) { return (int)((a + b - 1) / b); }

extern "C" void kernel_launch(void* const* d_in, const int* in_sizes, int n_in,
                              void* d_out, int out_size, void* d_ws, size_t ws_size,
                              hipStream_t stream) {
  (void)in_sizes; (void)n_in; (void)out_size; (void)ws_size;

  const float* x     = (const float*)d_in[0];   // [4096, 512]
  const float* nfr   = (const float*)d_in[1];   // [65536, 128]
  const int*   eidx  = (const int*)d_in[2];     // [2, E]
  const int*   pairs = (const int*)d_in[3];     // [P, 2] interleaved (node, hedge)
  const float* Wg1 = (const float*)d_in[4],  *bg1 = (const float*)d_in[5];
  const float* Wg2 = (const float*)d_in[6],  *bg2 = (const float*)d_in[7];
  const float* Wh1 = (const float*)d_in[8],  *bh1 = (const float*)d_in[9];
  const float* Wh2 = (const float*)d_in[10], *bh2 = (const float*)d_in[11];
  const float* Wm  = (const float*)d_in[12], *bm  = (const float*)d_in[13];
  const float* Wm2 = (const float*)d_in[14], *bm2 = (const float*)d_in[15];
  const float* Wsg = (const float*)d_in[16], *bsg = (const float*)d_in[17];
  const float* Wtg = (const float*)d_in[18], *btg = (const float*)d_in[19];
  float* out = (float*)d_out;

  const int* src = eidx;
  const int* dst = eidx + E_TOT;

  // workspace layout (floats)
  const long NN = (long)NTOT * EMB;            // 8,388,608
  float* ws    = (float*)d_ws;
  float* Hbuf  = ws;                           // [N,128] current node features
  float* Tbuf  = Hbuf + NN;                    // [N,128] gemm temp / kbuf
  float* Abuf  = Tbuf + NN;                    // [N,128] aggregation temp
  float* EFb   = Abuf + NN;                    // [N_HE,128]
  float* doutr = EFb + (long)N_HE * EMB;       // [N] rsqrt(out-degree)
  float* dinr  = doutr + NTOT;                 // [N] rsqrt(in-degree)
  float* Dinv  = dinr + NTOT;                  // [N] 1/hyper-degree
  float* Binv  = Dinv + NTOT;                  // [N_HE] 1/hyperedge-size
  float* Qbuf  = Binv + N_HE;                  // [4096,128]
  float* Hatt  = Qbuf + (long)ROWS_X * EMB;    // [4096,128]
  float* Cat   = Hatt + (long)ROWS_X * EMB;    // [4096,640]
  float* CWb   = Cat + (long)ROWS_X * CATD;    // [128] projected input col-bias

  // ---- degrees ----
  {
    long ndeg = 3L * NTOT + N_HE;
    zero_f32<<<cdiv(ndeg, 256), 256, 0, stream>>>(doutr, ndeg);
    deg_edges<<<E_TOT / 256, 256, 0, stream>>>(src, dst, doutr, dinr, E_TOT);
    deg_pairs<<<P_TOT / 256, 256, 0, stream>>>(pairs, Dinv, Binv, P_TOT);
    rsqrt_clamp<<<NTOT / 256, 256, 0, stream>>>(doutr, NTOT);
    rsqrt_clamp<<<NTOT / 256, 256, 0, stream>>>(dinr, NTOT);
    invert_pos<<<NTOT / 256, 256, 0, stream>>>(Dinv, NTOT);
    invert_pos<<<N_HE / 256, 256, 0, stream>>>(Binv, N_HE);
  }

  // ---- 2 GNN layers ----
  for (int layer = 0; layer < 2; ++layer) {
    const float* Xin = (layer == 0) ? nfr : Hbuf;
    const float* Wg = (layer == 0) ? Wg1 : Wg2;
    const float* bg = (layer == 0) ? bg1 : bg2;
    const float* Wh = (layer == 0) ? Wh1 : Wh2;
    const float* bh = (layer == 0) ? bh1 : bh2;

    // t = diag(rsqrt(dout)) * (Xin @ Wg)
    gemm128<<<NTOT / 128, 256, 0, stream>>>(Xin, EMB, NTOT, EMB, Wg,
                                            doutr, nullptr, nullptr,
                                            Tbuf, EMB, 0, 0);
    // agg[dst] += t[src]
    zero_f32<<<cdiv(NN, 256), 256, 0, stream>>>(Abuf, NN);
    scatter_edges<<<E_TOT / 8, 256, 0, stream>>>(Tbuf, Abuf, src, dst, E_TOT);
    // t2 = (agg * rsqrt(din) + bg) @ Wh  ==  diag(rsqrt(din))*(agg@Wh) + (bg@Wh)
    colbias_proj<<<1, 128, 0, stream>>>(bg, Wh, EMB, CWb);
    gemm128<<<NTOT / 128, 256, 0, stream>>>(Abuf, EMB, NTOT, EMB, Wh,
                                            dinr, CWb, nullptr,
                                            Tbuf, EMB, 0, 0);
    // ef[pe] += t2[pn]
    zero_f32<<<cdiv((long)N_HE * EMB, 256), 256, 0, stream>>>(EFb, (long)N_HE * EMB);
    scatter_pairs_fwd<<<P_TOT / 8, 256, 0, stream>>>(Tbuf, EFb, pairs, P_TOT);
    // hagg[pn] += Binv[pe] * ef[pe]
    zero_f32<<<cdiv(NN, 256), 256, 0, stream>>>(Abuf, NN);
    scatter_pairs_bwd<<<P_TOT / 8, 256, 0, stream>>>(EFb, Binv, Abuf, pairs, P_TOT);
    // h = relu(hagg * Dinv + bh) + nfr
    finalize_h<<<cdiv(NN, 256), 256, 0, stream>>>(Abuf, Dinv, bh, nfr, Hbuf, NN);
  }

  // ---- readout ----
  // q = relu(x @ Wm + bm)
  gemm128<<<ROWS_X / 128, 256, 0, stream>>>(x, ENCD, ROWS_X, ENCD, Wm,
                                            nullptr, nullptr, bm, Qbuf, EMB, 0, 1);
  // k = relu(h @ Wm2 + bm2)
  gemm128<<<NTOT / 128, 256, 0, stream>>>(Hbuf, EMB, NTOT, EMB, Wm2,
                                          nullptr, nullptr, bm2, Tbuf, EMB, 0, 1);
  // H = softmax(q k^T) g   (flash-style, per-batch)
  attn_kernel<<<ROWS_X / 64, 128, 0, stream>>>(Qbuf, Tbuf, Hbuf, Hatt);
  // cat = [x | H]
  concat_xh<<<cdiv((long)ROWS_X * CATD, 256), 256, 0, stream>>>(x, Hatt, Cat);
  // out[:,0:128] = sigmoid(cat @ Ws + bs); out[:,128:256] = tanh(cat @ Wt + bt)
  gemm128<<<ROWS_X / 128, 256, 0, stream>>>(Cat, CATD, ROWS_X, CATD, Wsg,
                                            nullptr, nullptr, bsg, out, OUTD, 0, 2);
  gemm128<<<ROWS_X / 128, 256, 0, stream>>>(Cat, CATD, ROWS_X, CATD, Wtg,
                                            nullptr, nullptr, btg, out, OUTD, 128, 3);
}